// TVNet_Scale_10608569221404
// MI455X (gfx1250) — compile-verified
//
#include <hip/hip_runtime.h>
#include <math.h>

// ---------------- problem constants (from reference) ----------------
#define EPSF   1e-12f
#define THETA  0.3f
#define L_T    (0.15f * 0.3f)     // LBDA * THETA
#define TAUT   (0.25f / 0.3f)     // TAU / THETA
static constexpr int Bn = 16, Hn = 512, Wn = 512;
static constexpr int NPIX = Bn * Hn * Wn;
static constexpr int TX = 32, TY = 8;   // 256 threads = 8 wave32 per block

// ---------------- CDNA5 async global->LDS helpers ----------------
__device__ __forceinline__ void async_load_f32(void* lds_dst, const float* src) {
    // low 32 bits of a generic LDS address == wave-relative LDS byte offset
    unsigned lds = (unsigned)(uintptr_t)lds_dst;
    asm volatile("global_load_async_to_lds_b32 %0, %1, off"
                 :: "v"(lds), "v"(src) : "memory");
}
__device__ __forceinline__ void async_wait0() {
    asm volatile("s_wait_asynccnt 0" ::: "memory");
}

// ---------------- kernel 1: centered gradient of x2 + seed u ----------------
// grid: (Wn/256, Hn, Bn), block: 256  -> no div/mod index math
__global__ void __launch_bounds__(256)
k_grad_init(const float* __restrict__ x2,
            const float* __restrict__ u1_in, const float* __restrict__ u2_in,
            float* __restrict__ gx, float* __restrict__ gy,
            float* __restrict__ u1, float* __restrict__ u2)
{
    int x = blockIdx.x * 256 + threadIdx.x;
    int y = blockIdx.y;
    int b = blockIdx.z;
    int idx = (b * Hn + y) * Wn + x;
    int rowbase = idx - x;
    int imgbase = rowbase - y * Wn;
    int xm = x > 0 ? x - 1 : 0;
    int xp = x < Wn - 1 ? x + 1 : Wn - 1;
    int ym = y > 0 ? y - 1 : 0;
    int yp = y < Hn - 1 ? y + 1 : Hn - 1;
    gx[idx] = 0.5f * (x2[rowbase + xp] - x2[rowbase + xm]);
    gy[idx] = 0.5f * (x2[imgbase + yp * Wn + x] - x2[imgbase + ym * Wn + x]);
    u1[idx] = u1_in[idx];
    u2[idx] = u2_in[idx];
}

// ---------------- kernel 2: bilinear warp of x2, gx, gy at (u1,u2) ----------------
// grid: (Wn/256, Hn, Bn), block: 256
__global__ void __launch_bounds__(256)
k_warp(const float* __restrict__ x1, const float* __restrict__ x2,
       const float* __restrict__ gx, const float* __restrict__ gy,
       const float* __restrict__ u1, const float* __restrict__ u2,
       float* __restrict__ dxw, float* __restrict__ dyw,
       float* __restrict__ rho_c)
{
    int x = blockIdx.x * 256 + threadIdx.x;
    int y = blockIdx.y;
    int b = blockIdx.z;
    int base = b * Hn * Wn;
    int idx = base + y * Wn + x;
    float U1 = u1[idx], U2 = u2[idx];
    // px = (gx[x] + u*2/W + 1) * W/2  ==  x * W/(W-1) + u
    float px = (float)x * ((float)Wn / (float)(Wn - 1)) + U1;
    float py = (float)y * ((float)Hn / (float)(Hn - 1)) + U2;
    int x0 = (int)floorf(px), y0 = (int)floorf(py);
    int x0c = min(max(x0, 0), Wn - 1);
    int x1c = min(max(x0 + 1, 0), Wn - 1);
    int y0c = min(max(y0, 0), Hn - 1);
    int y1c = min(max(y0 + 1, 0), Hn - 1);
    float x0f = (float)x0c, x1f = (float)x1c, y0f = (float)y0c, y1f = (float)y1c;
    float wa = (x1f - px) * (y1f - py);
    float wb = (x1f - px) * (py - y0f);
    float wc = (px - x0f) * (y1f - py);
    float wd = (px - x0f) * (py - y0f);
    int ia = base + y0c * Wn + x0c;
    int ib = base + y1c * Wn + x0c;
    int ic = base + y0c * Wn + x1c;
    int id = base + y1c * Wn + x1c;
    float x2w = wa * x2[ia] + wb * x2[ib] + wc * x2[ic] + wd * x2[id];
    float dxv = wa * gx[ia] + wb * gx[ib] + wc * gx[ic] + wd * gx[id];
    float dyv = wa * gy[ia] + wb * gy[ib] + wc * gy[ic] + wd * gy[id];
    dxw[idx]   = dxv;
    dyw[idx]   = dyv;
    rho_c[idx] = x2w - dxv * U1 - dyv * U2 - x1[idx];
}

// ---------------- fused per-iteration kernel ----------------
// Phase 1: u-update on extended (TX+1)x(TY+1) tile (redundant halo compute),
//          divergence of p from a (TX+2)x(TY+2) async-staged LDS tile.
// Phase 2: p-update on interior tile; forward gradient of u_new from LDS,
//          p(own pixel) reused from the staged LDS tile (no global re-read).
// u is ping-ponged (ur -> uw) to avoid inter-block halo races.
__global__ void __launch_bounds__(TX * TY)
k_iter(const float* __restrict__ ur1, const float* __restrict__ ur2,
       float* __restrict__ uw1, float* __restrict__ uw2,
       float* __restrict__ p11, float* __restrict__ p12,
       float* __restrict__ p21, float* __restrict__ p22,
       const float* __restrict__ dxw, const float* __restrict__ dyw,
       const float* __restrict__ rho_c,
       float* __restrict__ rho_out, int write_rho)
{
    __shared__ float s11[TY + 2][TX + 2];
    __shared__ float s12[TY + 2][TX + 2];
    __shared__ float s21[TY + 2][TX + 2];
    __shared__ float s22[TY + 2][TX + 2];
    __shared__ float un1[TY + 1][TX + 1];
    __shared__ float un2[TY + 1][TX + 1];

    int b  = blockIdx.z;
    int x0 = blockIdx.x * TX, y0 = blockIdx.y * TY;
    int tid = threadIdx.y * TX + threadIdx.x;
    int base = b * Hn * Wn;

    // stage p tile [x0-1 .. x0+TX] x [y0-1 .. y0+TY] (clamped) via async->LDS
    for (int i = tid; i < (TX + 2) * (TY + 2); i += TX * TY) {
        int sx = i % (TX + 2), sy = i / (TX + 2);
        int gxi = min(max(x0 - 1 + sx, 0), Wn - 1);
        int gyi = min(max(y0 - 1 + sy, 0), Hn - 1);
        int g = base + gyi * Wn + gxi;
        async_load_f32(&s11[sy][sx], p11 + g);
        async_load_f32(&s12[sy][sx], p12 + g);
        async_load_f32(&s21[sy][sx], p21 + g);
        async_load_f32(&s22[sy][sx], p22 + g);
    }
    async_wait0();
    __syncthreads();

    // phase 1: u-update over extended region
    for (int i = tid; i < (TX + 1) * (TY + 1); i += TX * TY) {
        int ex = i % (TX + 1), ey = i / (TX + 1);
        int x = x0 + ex, y = y0 + ey;
        if (x < Wn && y < Hn) {
            int idx = base + y * Wn + x;
            int sx = ex + 1, sy = ey + 1;     // p tile coords
            float c11 = s11[sy][sx], l11 = s11[sy][sx - 1];
            float c12 = s12[sy][sx], t12 = s12[sy - 1][sx];
            float c21 = s21[sy][sx], l21 = s21[sy][sx - 1];
            float c22 = s22[sy][sx], t22 = s22[sy - 1][sx];
            float dx1 = (x == 0) ? c11 : ((x == Wn - 1) ? -l11 : c11 - l11);
            float dy1 = (y == 0) ? c12 : ((y == Hn - 1) ? -t12 : c12 - t12);
            float dx2 = (x == 0) ? c21 : ((x == Wn - 1) ? -l21 : c21 - l21);
            float dy2 = (y == 0) ? c22 : ((y == Hn - 1) ? -t22 : c22 - t22);

            float U1 = ur1[idx], U2 = ur2[idx];
            float DX = dxw[idx], DY = dyw[idx], RC = rho_c[idx];
            float G  = DX * DX + DY * DY + EPSF;   // grad recomputed: 2 VALU ops vs 16.8MB/iter
            float rho = RC + DX * U1 + DY * U2 + EPSF;
            float th  = L_T * G;
            float s   = (rho < -th) ? L_T
                      : ((rho > th) ? -L_T
                      : ((G > EPSF) ? (-rho / G) : 0.0f));
            float nu1 = s * DX + U1 + THETA * (dx1 + dy1);
            float nu2 = s * DY + U2 + THETA * (dx2 + dy2);
            un1[ey][ex] = nu1;
            un2[ey][ex] = nu2;
            if (ex < TX && ey < TY) {
                uw1[idx] = nu1;
                uw2[idx] = nu2;
                if (write_rho) rho_out[idx] = rho;
            }
        }
    }
    __syncthreads();

    // phase 2: p-update at interior pixel
    {
        int ex = threadIdx.x, ey = threadIdx.y;
        int x = x0 + ex, y = y0 + ey;
        int idx = base + y * Wn + x;
        float U1 = un1[ey][ex], U2 = un2[ey][ex];
        float u1x = (x < Wn - 1) ? un1[ey][ex + 1] - U1 : 0.0f;
        float u1y = (y < Hn - 1) ? un1[ey + 1][ex] - U1 : 0.0f;
        float u2x = (x < Wn - 1) ? un2[ey][ex + 1] - U2 : 0.0f;
        float u2y = (y < Hn - 1) ? un2[ey + 1][ex] - U2 : 0.0f;
        float ng1 = 1.0f + TAUT * sqrtf(u1x * u1x + u1y * u1y + EPSF);
        float ng2 = 1.0f + TAUT * sqrtf(u2x * u2x + u2y * u2y + EPSF);
        float r1 = 1.0f / ng1, r2 = 1.0f / ng2;
        // p(own pixel) comes from the staged LDS tile — no global re-read
        p11[idx] = (s11[ey + 1][ex + 1] + TAUT * u1x) * r1;
        p12[idx] = (s12[ey + 1][ex + 1] + TAUT * u1y) * r1;
        p21[idx] = (s21[ey + 1][ex + 1] + TAUT * u2x) * r2;
        p22[idx] = (s22[ey + 1][ex + 1] + TAUT * u2y) * r2;
    }
}

// ---------------- host driver ----------------
extern "C" void kernel_launch(void* const* d_in, const int* in_sizes, int n_in,
                              void* d_out, int out_size, void* d_ws, size_t ws_size,
                              hipStream_t stream)
{
    const float* x1    = (const float*)d_in[0];
    const float* x2    = (const float*)d_in[1];
    const float* u1_in = (const float*)d_in[2];
    const float* u2_in = (const float*)d_in[3];

    float* out = (float*)d_out;   // outputs: (u1, u2, rho) flat-concatenated
    float* uA1 = out;
    float* uA2 = out + NPIX;
    float* rho = out + 2 * NPIX;

    float* ws    = (float*)d_ws;  // needs 11 * NPIX * 4B = 185 MB scratch
    float* gx    = ws + 0  * (size_t)NPIX;
    float* gy    = ws + 1  * (size_t)NPIX;
    float* dxw   = ws + 2  * (size_t)NPIX;
    float* dyw   = ws + 3  * (size_t)NPIX;
    float* rho_c = ws + 4  * (size_t)NPIX;
    float* p11   = ws + 5  * (size_t)NPIX;
    float* p12   = ws + 6  * (size_t)NPIX;
    float* p21   = ws + 7  * (size_t)NPIX;
    float* p22   = ws + 8  * (size_t)NPIX;
    float* uB1   = ws + 9  * (size_t)NPIX;
    float* uB2   = ws + 10 * (size_t)NPIX;

    const dim3 pgrid(Wn / 256, Hn, Bn);       // pointwise kernels
    const dim3 tgrid(Wn / TX, Hn / TY, Bn);   // tiled iteration kernel
    const dim3 tblock(TX, TY, 1);

    k_grad_init<<<pgrid, 256, 0, stream>>>(x2, u1_in, u2_in, gx, gy, uA1, uA2);
    // p11..p22 are contiguous: zero them (dual variables start at 0 every call)
    hipMemsetAsync(p11, 0, (size_t)4 * NPIX * sizeof(float), stream);

    for (int w = 0; w < 5; ++w) {
        // u lives in buffer A at every warp boundary (30 iterations = even)
        k_warp<<<pgrid, 256, 0, stream>>>(x1, x2, gx, gy, uA1, uA2,
                                          dxw, dyw, rho_c);
        for (int it = 0; it < 30; ++it) {
            int wr = (w == 4 && it == 29) ? 1 : 0;
            const float* r1p = (it & 1) ? uB1 : uA1;
            const float* r2p = (it & 1) ? uB2 : uA2;
            float*       w1p = (it & 1) ? uA1 : uB1;
            float*       w2p = (it & 1) ? uA2 : uB2;
            k_iter<<<tgrid, tblock, 0, stream>>>(r1p, r2p, w1p, w2p,
                                                 p11, p12, p21, p22,
                                                 dxw, dyw, rho_c, rho, wr);
        }
    }
}